// BN2d_54331336294636
// MI455X (gfx1250) — compile-verified
//
#include <hip/hip_runtime.h>
#include <math.h>

// Problem geometry (fixed by the reference): N=64, C=128, H=W=56.
#define CCH     128
#define HW4     784              // 56*56/4 float4 per (n, c) plane
#define SPLIT1  7                // blocks per channel, small-stats pass (16*784/1792)
#define SPLIT2  28               // blocks per channel, full passes      (64*784/1792)
#define CHUNK4  1792             // float4 per block = 256 threads * 7
#define M_SMALL 50176.0f         // 16*56*56
#define THRES   3.0f

typedef __attribute__((ext_vector_type(2))) float v2f;
typedef __attribute__((ext_vector_type(4))) float v4f;
typedef __attribute__((ext_vector_type(8))) float v8f;

// Sum of the fp32 values of threads with tid % 64 < 32 (blockDim.x == 256).
// LDS folds sm[tid + 64j], j < 4 (128 of the 256) -> 32 (one partial per lane of wave 0), then a single
// V_WMMA_F32_16X16X4_F32 with A = ones collapses 32 -> 16 in one matrix op
// (D[m][n] = x[n] + x[n+16], identical in every row), and 4 shfl_xor steps
// finish. Full f32 precision throughout. Result valid in wave 0.
__device__ __forceinline__ float block_reduce_256(float p, float* sm) {
  const int tid = threadIdx.x;
  sm[tid] = p;
  __syncthreads();
  float total = 0.0f;
  if (tid < 32) {                      // wave 0: EXEC all-ones inside this branch
    float x0 = sm[tid] + sm[tid + 64] + sm[tid + 128] + sm[tid + 192];
#if __has_builtin(__builtin_amdgcn_wmma_f32_16x16x4_f32)
    v2f a; a[0] = 1.0f; a[1] = 1.0f;   // A = ones(16x4)
    v2f b; b[0] = x0;   b[1] = 0.0f;   // B rows 0/2 carry the 32 partials
    v8f c = {0.f,0.f,0.f,0.f,0.f,0.f,0.f,0.f};
    c = __builtin_amdgcn_wmma_f32_16x16x4_f32(false, a, false, b,
                                              (short)0, c, false, false);
    float d = c[0];                    // lane L holds x0[L&15] + x0[(L&15)+16]
#else
    float d = x0 + __shfl_xor(x0, 16, 32);
#endif
    d += __shfl_xor(d, 8, 32);
    d += __shfl_xor(d, 4, 32);
    d += __shfl_xor(d, 2, 32);
    d += __shfl_xor(d, 1, 32);
    total = d;
  }
  __syncthreads();                     // LDS reused by subsequent reductions
  return total;
}

// Pass 1: per-channel sum / sumsq over the first 16 batches.
__global__ void __launch_bounds__(256)
bn_small_stats(const float* __restrict__ x, float* __restrict__ ws) {
  __shared__ float sm[256];
  const int c     = blockIdx.x / SPLIT1;
  const int chunk = blockIdx.x % SPLIT1;
  const v4f* __restrict__ x4 = (const v4f*)x;

  float s = 0.0f, sq = 0.0f;
  int i = chunk * CHUNK4 + threadIdx.x;
#pragma unroll
  for (int k = 0; k < 7; ++k, i += 256) {
    const int n  = i / HW4;            // 0..15
    const int hw = i - n * HW4;
    const v4f v = x4[(n * CCH + c) * HW4 + hw];
    s += v[0] + v[1] + v[2] + v[3];
#pragma unroll
    for (int j = 0; j < 4; ++j) sq = fmaf(v[j], v[j], sq);
  }
  const float ts = block_reduce_256(s,  sm);
  const float tq = block_reduce_256(sq, sm);
  if (threadIdx.x == 0) {
    atomicAdd(&ws[c],        ts);
    atomicAdd(&ws[CCH + c],  tq);
  }
}

// Pass 2: per-channel masked cnt / sum / sumsq over the full tensor.
__global__ void __launch_bounds__(256)
bn_masked_sums(const float* __restrict__ x, float* __restrict__ ws) {
  __shared__ float sm[256];
  const int c     = blockIdx.x / SPLIT2;
  const int chunk = blockIdx.x % SPLIT2;

  const float sumA  = ws[c];
  const float sqA   = ws[CCH + c];
  const float mean  = sumA * (1.0f / M_SMALL);
  const float var   = fmaf(-sumA, mean, sqA) * (1.0f / (M_SMALL - 1.0f)); // ddof=1
  const float rstd  = rsqrtf(var + 1e-10f);

  const v4f* __restrict__ x4 = (const v4f*)x;
  float cnt = 0.0f, s1 = 0.0f, s2 = 0.0f;
  int i = chunk * CHUNK4 + threadIdx.x;
#pragma unroll
  for (int k = 0; k < 7; ++k, i += 256) {
    const int n  = i / HW4;            // 0..63
    const int hw = i - n * HW4;
    const v4f v = x4[(n * CCH + c) * HW4 + hw];
#pragma unroll
    for (int j = 0; j < 4; ++j) {
      const float e   = v[j];
      const float z   = (e - mean) * rstd;
      const float sel = (fabsf(z) < THRES) ? 1.0f : 0.0f;
      cnt += sel;
      s1  = fmaf(e,     sel, s1);
      s2  = fmaf(e * e, sel, s2);
    }
  }
  const float tc = block_reduce_256(cnt, sm);
  const float t1 = block_reduce_256(s1,  sm);
  const float t2 = block_reduce_256(s2,  sm);
  if (threadIdx.x == 0) {
    atomicAdd(&ws[2 * CCH + c], tc);
    atomicAdd(&ws[3 * CCH + c], t1);
    atomicAdd(&ws[4 * CCH + c], t2);
  }
}

// Pass 3: elementwise affine output. x should hit L2 (103MB < 192MB) from
// pass 2; use non-temporal stores so the output stream doesn't evict it.
__global__ void __launch_bounds__(256)
bn_output(const float* __restrict__ x, const float* __restrict__ gamma,
          const float* __restrict__ beta, const float* __restrict__ ws,
          float* __restrict__ out) {
  const int c     = blockIdx.x / SPLIT2;
  const int chunk = blockIdx.x % SPLIT2;

  const float cnt   = ws[2 * CCH + c];
  const float s1    = ws[3 * CCH + c];
  const float s2    = ws[4 * CCH + c];
  const float dmean = s1 / cnt;
  const float dvar  = fmaf(-dmean, dmean, s2 / cnt);   // s2/cnt - dmean^2
  const float sc    = gamma[c] * rsqrtf(dvar + 1e-20f);
  const float sh    = fmaf(-sc, dmean, beta[c]);

  const v4f* __restrict__ x4 = (const v4f*)x;
  v4f* __restrict__ o4 = (v4f*)out;
  int i = chunk * CHUNK4 + threadIdx.x;
#pragma unroll
  for (int k = 0; k < 7; ++k, i += 256) {
    const int n   = i / HW4;
    const int hw  = i - n * HW4;
    const int idx = (n * CCH + c) * HW4 + hw;
    v4f v = x4[idx];
#pragma unroll
    for (int j = 0; j < 4; ++j) v[j] = fmaf(v[j], sc, sh);
    __builtin_nontemporal_store(v, &o4[idx]);
  }
}

extern "C" void kernel_launch(void* const* d_in, const int* in_sizes, int n_in,
                              void* d_out, int out_size, void* d_ws, size_t ws_size,
                              hipStream_t stream) {
  const float* x     = (const float*)d_in[0];   // (64,128,56,56) fp32
  const float* gamma = (const float*)d_in[1];   // (1,128,1,1)
  const float* beta  = (const float*)d_in[2];   // (1,128,1,1)
  float* out = (float*)d_out;
  float* ws  = (float*)d_ws;

  // ws: [sumA | sumsqA | cnt | s1 | s2], 128 floats each. Must be zeroed
  // every launch (deterministic under graph replay). Memset node is capture-safe.
  (void)hipMemsetAsync(d_ws, 0, 5 * CCH * sizeof(float), stream);

  bn_small_stats<<<CCH * SPLIT1, 256, 0, stream>>>(x, ws);
  bn_masked_sums<<<CCH * SPLIT2, 256, 0, stream>>>(x, ws);
  bn_output     <<<CCH * SPLIT2, 256, 0, stream>>>(x, gamma, beta, ws, out);
}